// CrossCell_65910568124539
// MI455X (gfx1250) — compile-verified
//
#include <hip/hip_runtime.h>
#include <hip/hip_bf16.h>

typedef __attribute__((ext_vector_type(2))) float v2f;
typedef __attribute__((ext_vector_type(8))) float v8f;

#define IN_F 128
#define OUT_F 128
#define LDS_STRIDE 132   // 128 + 4 pad: lane L hits bank (c + 4L) % 64 -> conflict-free

// ---------------------------------------------------------------------------
// Kernel 1: fused dual GEMM.
//   h1   = X @ W1 + b1              -> written to d_out (the "+ h1" term)
//   hsum = h1 + (X*X) @ W2 + b2     -> written to workspace (SpMM input)
// One block = 16 node rows; 8 waves each compute one 16x16 output tile via
// V_WMMA_F32_16X16X4_F32, K-loop of 32 steps, 2 WMMAs per step.
// ---------------------------------------------------------------------------
__global__ __launch_bounds__(256) void gemm_h_kernel(
    const float* __restrict__ features,
    const float* __restrict__ W1, const float* __restrict__ b1,
    const float* __restrict__ W2, const float* __restrict__ b2,
    float* __restrict__ out_h1, float* __restrict__ out_hsum, int N)
{
    __shared__ float ldsA[16 * LDS_STRIDE];
    const int tid  = threadIdx.x;
    const int row0 = blockIdx.x * 16;

    // Cooperative, coalesced stage of the 16x128 A tile into LDS.
    for (int i = tid; i < 16 * IN_F; i += 256) {
        const int r = i >> 7;          // /128
        const int c = i & (IN_F - 1);
        const int grow = row0 + r;
        ldsA[r * LDS_STRIDE + c] = (grow < N) ? features[(size_t)grow * IN_F + c] : 0.0f;
    }
    __syncthreads();

    const int wave = tid >> 5;
    const int lane = tid & 31;
    const int m    = lane & 15;            // M index within tile (A) / N index (B,C)
    const int kofs = (lane >> 4) << 1;     // 0 for lanes 0-15, 2 for lanes 16-31
    const int col  = wave * 16 + m;        // output column this lane owns in B/C/D

    v8f c1 = {};   // accumulator for X @ W1
    v8f c2 = {};   // accumulator for (X*X) @ W2

    #pragma unroll 4
    for (int kk = 0; kk < IN_F / 4; ++kk) {
        const int kbase = kk * 4 + kofs;
        // A fragment (16x4 f32): lane holds A[m][kbase], A[m][kbase+1]
        v2f a;
        a.x = ldsA[m * LDS_STRIDE + kbase];
        a.y = ldsA[m * LDS_STRIDE + kbase + 1];
        v2f a2 = a * a;                    // squared-feature fragment, in-register
        // B fragments (4x16 f32): vgpr j holds W[(kbase+j)][col]
        const float* w1p = W1 + (size_t)kbase * OUT_F + col;
        const float* w2p = W2 + (size_t)kbase * OUT_F + col;
        v2f bb1; bb1.x = w1p[0]; bb1.y = w1p[OUT_F];
        v2f bb2; bb2.x = w2p[0]; bb2.y = w2p[OUT_F];

        c1 = __builtin_amdgcn_wmma_f32_16x16x4_f32(
                 false, a,  false, bb1, (short)0, c1, false, false);
        c2 = __builtin_amdgcn_wmma_f32_16x16x4_f32(
                 false, a2, false, bb2, (short)0, c2, false, false);
    }

    // C/D layout: VGPR v, lanes 0-15 -> M=v ; lanes 16-31 -> M=v+8 (N = lane&15)
    const float bias1 = b1[col];
    const float bias2 = b2[col];
    const int   rbase = row0 + ((lane >> 4) << 3);

    if (row0 + 16 <= N) {
        // Fast path: whole tile in range (always taken when N % 16 == 0).
        #pragma unroll
        for (int v = 0; v < 8; ++v) {
            const float h1v = c1[v] + bias1;
            const float hs  = h1v + c2[v] + bias2;
            const size_t idx = (size_t)(rbase + v) * OUT_F + col;
            out_h1[idx]   = h1v;
            out_hsum[idx] = hs;
        }
    } else {
        #pragma unroll
        for (int v = 0; v < 8; ++v) {
            const int row = rbase + v;
            if (row < N) {
                const float h1v = c1[v] + bias1;
                const float hs  = h1v + c2[v] + bias2;
                const size_t idx = (size_t)row * OUT_F + col;
                out_h1[idx]   = h1v;
                out_hsum[idx] = hs;
            }
        }
    }
}

// ---------------------------------------------------------------------------
// Guaranteed-native non-returning FP32 L2 atomic, device scope.
// Avoids any chance of a CAS-loop fallback from the generic atomic builtins.
// ---------------------------------------------------------------------------
__device__ __forceinline__ void atomic_add_f32_dev(float* p, float v) {
    asm volatile("global_atomic_add_f32 %0, %1, off scope:SCOPE_DEV"
                 :: "v"(p), "v"(v) : "memory");
}

// ---------------------------------------------------------------------------
// Kernel 2: fused SpMM scatter:  out[row] += val * hsum[col]   (atomics)
// One wave per edge: lane l moves 4 consecutive floats (global_load_b128 +
// 4x global_atomic_add_f32, non-returning). Both hsum and out fit in L2.
// Edge indices are forced wave-uniform so they become scalar s_loads.
// ---------------------------------------------------------------------------
__global__ __launch_bounds__(256) void spmm_scatter_kernel(
    const int*   __restrict__ erows,
    const int*   __restrict__ ecols,
    const float* __restrict__ evals,
    const float* __restrict__ h,
    float*       __restrict__ out, int E)
{
    const int lane = threadIdx.x & 31;
    const int wave = (blockIdx.x * blockDim.x + threadIdx.x) >> 5;
    const int nw   = (gridDim.x * blockDim.x) >> 5;

    for (int e0 = wave; e0 < E; e0 += nw) {
        const int   e = __builtin_amdgcn_readfirstlane(e0);  // provably uniform -> s_load
        const int   r = erows[e];
        const int   c = ecols[e];
        const float v = evals[e];
        const float4 g = ((const float4*)(h + (size_t)c * OUT_F))[lane];
        float* dst = out + (size_t)r * OUT_F + (size_t)(lane * 4);
        atomic_add_f32_dev(dst + 0, g.x * v);
        atomic_add_f32_dev(dst + 1, g.y * v);
        atomic_add_f32_dev(dst + 2, g.z * v);
        atomic_add_f32_dev(dst + 3, g.w * v);
    }
}

extern "C" void kernel_launch(void* const* d_in, const int* in_sizes, int n_in,
                              void* d_out, int out_size, void* d_ws, size_t ws_size,
                              hipStream_t stream)
{
    const float* features = (const float*)d_in[0];
    const int*   erows    = (const int*)  d_in[1];
    const int*   ecols    = (const int*)  d_in[2];
    const float* evals    = (const float*)d_in[3];
    const float* W1       = (const float*)d_in[4];
    const float* b1       = (const float*)d_in[5];
    const float* W2       = (const float*)d_in[6];
    const float* b2       = (const float*)d_in[7];

    float* out  = (float*)d_out;
    float* hbuf = (float*)d_ws;    // hsum: N * 128 floats (51.2 MB)

    const int N = in_sizes[0] / IN_F;
    const int E = in_sizes[1];

    const int nblocks = (N + 15) / 16;
    gemm_h_kernel<<<nblocks, 256, 0, stream>>>(features, W1, b1, W2, b2, out, hbuf, N);
    spmm_scatter_kernel<<<4096, 256, 0, stream>>>(erows, ecols, evals, hbuf, out, E);
}